// WaveletTransform3D_57423712748251
// MI455X (gfx1250) — compile-verified
//
#include <hip/hip_runtime.h>

// 3D Haar LL band: out[b,c,dz,hy,wx] = 2^-1.5 * sum of the 2x2x2 input block.
// x: (2,16,128,128,128) f32  ->  out: (2,16,64,64,64) f32
//
// Bandwidth-bound (256 MiB in + 32 MiB out, ~67 MFLOP; ~12.4 us floor at 23.3 TB/s).
//  - lane L in 0..15 handles output voxel m=L (d-plane 2dz), lane L+16 the same
//    voxel's 2dz+1 plane; each lane W-pair-sums two float2 loads. That is exactly
//    the CDNA5 16x4 f32 WMMA A-layout (lanes 0-15: K=0,1 / lanes 16-31: K=2,3).
//  - V_WMMA_F32_16X16X4_F32 with B = all 2^-1.5 finishes the 2x2x2 sum in f32.
//  - Phase structure: 8 NT loads first (max MLP), then 4 WMMAs + select trees,
//    then one predicated region with 4 NT stores (single exec round-trip).

typedef __attribute__((ext_vector_type(2))) float v2f;
typedef __attribute__((ext_vector_type(8))) float v8f;

#define HAAR_SCALE 0.35355339059327373f  // 2^-1.5

__global__ __launch_bounds__(256) void haar_ll3d_wmma(const float* __restrict__ x,
                                                      float* __restrict__ out) {
  const int tid  = threadIdx.x;
  const int lane = tid & 31;
  const int wave = tid >> 5;

  // One output row (b,c,dz,hy) of 64 voxels per wave. Rows total = 2*16*64*64 = 131072.
  const long rowId = (long)blockIdx.x * 8 + wave;
  const int  hy = (int)(rowId & 63);
  const long t1 = rowId >> 6;
  const int  dz = (int)(t1 & 63);
  const int  bc = (int)(t1 >> 6);          // b*16 + c (stride D*H*W)

  const int m  = lane & 15;                // voxel index within a 16-chunk
  const int dd = lane >> 4;                // which d-plane this lane loads (0/1)

  // Input row base for this lane: (bc, 2dz+dd, 2hy, 0)
  const size_t inRow   = (((size_t)(bc * 128 + 2 * dz + dd)) * 128 + (size_t)(2 * hy)) * 128;
  const size_t outBase = (size_t)rowId * 64;

  const v2f bmat = {HAAR_SCALE, HAAR_SCALE};
  const int k = lane & 7;
  const bool doStore = ((lane & 8) == 0);
  const size_t laneOut = outBase + (size_t)(((lane >> 4) << 3) + k);

  // ---- Phase 1: issue all 8 streaming loads (2 aligned 128B segments each) ----
  v2f r0[4], r1[4];
  const float* p = x + inRow + (size_t)(2 * m);
#pragma unroll
  for (int wg = 0; wg < 4; ++wg) {
    r0[wg] = __builtin_nontemporal_load((const v2f*)(p + 32 * wg));        // row (.., 2hy)
    r1[wg] = __builtin_nontemporal_load((const v2f*)(p + 32 * wg + 128));  // row (.., 2hy+1)
  }

  // ---- Phase 2: 4 WMMAs + voxel->lane routing, no exec changes ----
  float vals[4];
#pragma unroll
  for (int wg = 0; wg < 4; ++wg) {
    v2f a;
    a[0] = r0[wg][0] + r0[wg][1];   // K = 0 (lanes<16) / K = 2 (lanes>=16)
    a[1] = r1[wg][0] + r1[wg][1];   // K = 1 (lanes<16) / K = 3 (lanes>=16)
    v8f c = {};
    // D[m,n] = scale * sum_k A[m,k]  (all n)  == finished LL value of voxel m
    c = __builtin_amdgcn_wmma_f32_16x16x4_f32(false, a, false, bmat,
                                              (short)0, c, false, false);
    // Lanes 0-15 hold voxels 0..7 in c[0..7]; lanes 16-31 hold voxels 8..15.
    float v01 = (k & 1) ? c[1] : c[0];
    float v23 = (k & 1) ? c[3] : c[2];
    float v45 = (k & 1) ? c[5] : c[4];
    float v67 = (k & 1) ? c[7] : c[6];
    float v03 = (k & 2) ? v23 : v01;
    float v47 = (k & 2) ? v67 : v45;
    vals[wg] = (k & 4) ? v47 : v03;
  }

  // ---- Phase 3: one predicated region, 4 NT stores ----
  if (doStore) {
#pragma unroll
    for (int wg = 0; wg < 4; ++wg) {
      __builtin_nontemporal_store(vals[wg], out + laneOut + (size_t)(wg * 16));
    }
  }
}

extern "C" void kernel_launch(void* const* d_in, const int* in_sizes, int n_in,
                              void* d_out, int out_size, void* d_ws, size_t ws_size,
                              hipStream_t stream) {
  (void)in_sizes; (void)n_in; (void)out_size; (void)d_ws; (void)ws_size;
  const float* x = (const float*)d_in[0];
  float* out = (float*)d_out;
  // 131072 output rows, 8 waves (one row each) per 256-thread block.
  dim3 grid(16384), block(256);
  hipLaunchKernelGGL(haar_ll3d_wmma, grid, block, 0, stream, x, out);
}